// DetectionPostProcessV1_82884278879264
// MI455X (gfx1250) — compile-verified
//
#include <hip/hip_runtime.h>
#include <hip/hip_bf16.h>
#include <stdint.h>

#ifndef __has_builtin
#define __has_builtin(x) 0
#endif

// ---------------- problem constants (match reference) ----------------
#define BB     4
#define AAN    6
#define CCN    80
#define HHN    128
#define WWN    128
#define HWP    (HHN*WWN)        // 16384
#define CHN    (4+CCN)          // 84 channels per anchor
#define NNP    (AAN*HWP)        // 98304 anchors per batch
#define TT     1024             // sort threads per batch
#define CHUNK  (NNP/TT)         // 96 anchors per sort thread
#define KPRE   4095
#define KWRD   128              // ceil(4095/32)
#define POSTK  300
#define NEGV   (-(1<<30))
#define IMGF   1024.0f
#define NMSTHR 0.5f

// ---------------- gfx1250-specific helpers ----------------
#if defined(__HIP_DEVICE_COMPILE__) && defined(__gfx1250__)
#define GFX1250_PATH 1
typedef int v4i_t __attribute__((ext_vector_type(4)));
typedef __attribute__((address_space(1))) v4i_t gv4i_t;   // global int4
typedef __attribute__((address_space(3))) v4i_t lv4i_t;   // LDS int4
typedef __attribute__((address_space(3))) char lds_char_t;
#endif

// async copy 16B global -> LDS (per active lane); falls back to plain copy
__device__ __forceinline__ void async_copy_b128(const void* gsrc, void* ldst) {
#if defined(GFX1250_PATH)
#if __has_builtin(__builtin_amdgcn_global_load_async_to_lds_b128)
  __builtin_amdgcn_global_load_async_to_lds_b128((gv4i_t*)gsrc, (lv4i_t*)ldst, 0, 0);
#else
  unsigned lds_off = (unsigned)(unsigned long long)(lds_char_t*)ldst;
  unsigned long long gaddr = (unsigned long long)gsrc;
  asm volatile("global_load_async_to_lds_b128 %0, %1, off"
               :: "v"(lds_off), "v"(gaddr) : "memory");
#endif
#else
  const float* s = (const float*)gsrc;
  float* d = (float*)ldst;
  d[0] = s[0]; d[1] = s[1]; d[2] = s[2]; d[3] = s[3];
#endif
}

__device__ __forceinline__ void wait_async0() {
#if defined(GFX1250_PATH)
#if __has_builtin(__builtin_amdgcn_s_wait_asynccnt)
  __builtin_amdgcn_s_wait_asynccnt(0);
#else
  asm volatile("s_wait_asynccnt 0" ::: "memory");
#endif
#endif
}

// ---------------- kernel 1: class max + argmax (the big 132MB sweep) ----------------
// int4-vectorized: each thread owns 4 consecutive pixels; 80 b128 loads/thread,
// a full 512B cacheline-run per wave per channel.
__global__ void score_kernel(const int* __restrict__ data,
                             int* __restrict__ score, int* __restrict__ label) {
  int pix4 = blockIdx.x * 256 + threadIdx.x;          // 0..4095 (group of 4 pixels)
  int a    = blockIdx.y;
  int b    = blockIdx.z;
  const int4* cls = (const int4*)(data + (((size_t)(b*AAN + a)*CHN) + 4) * HWP) + pix4;
  int b0 = -0x40000000, b1 = b0, b2 = b0, b3 = b0;
  int l0 = 0, l1 = 0, l2 = 0, l3 = 0;
  #pragma unroll 4
  for (int c = 0; c < CCN; ++c) {
    if ((c & 15) == 0)                                // gfx1250 global_prefetch_b8
      __builtin_prefetch(cls + (size_t)(c + 16) * (HWP/4), 0, 0);
    int4 v = cls[(size_t)c * (HWP/4)];
    if (v.x > b0) { b0 = v.x; l0 = c; }
    if (v.y > b1) { b1 = v.y; l1 = c; }
    if (v.z > b2) { b2 = v.z; l2 = c; }
    if (v.w > b3) { b3 = v.w; l3 = c; }
  }
  size_t o4 = ((size_t)b*NNP + (size_t)a*HWP)/4 + pix4;
  ((int4*)score)[o4] = make_int4(b0, b1, b2, b3);
  ((int4*)label)[o4] = make_int4(l0, l1, l2, l3);     // CLASS_OFFSET == 0
}

// ---------------- kernel 2: per-thread stable histograms ----------------
__global__ void hist_kernel(const int* __restrict__ score, int* __restrict__ thist) {
  int b = blockIdx.x, t = threadIdx.x;                // 1024 threads
  for (int s = 0; s < 128; ++s)
    thist[((size_t)(b*128 + s))*TT + t] = 0;
  size_t base = (size_t)b*NNP + (size_t)t*CHUNK;
  for (int e = 0; e < CHUNK; ++e) {
    int s = score[base + e];
    if (s >= 1) {
      size_t idx = ((size_t)(b*128 + s))*TT + t;
      thist[idx] = thist[idx] + 1;                    // column exclusive to thread t
    }
  }
}

// ---------------- kernel 3: cross-thread exclusive scan per bin ----------------
__global__ void scan_kernel(int* __restrict__ thist) {
  __shared__ int total[128];
  __shared__ int basesh[128];
  int b = blockIdx.x, s = threadIdx.x;                // 128 threads, one per score bin
  int* col = thist + ((size_t)(b*128 + s))*TT;
  int sum = 0;
  for (int i = 0; i < TT; ++i) sum += col[i];
  total[s] = sum;
  __syncthreads();
  if (s == 0) {                                       // descending-score bases
    int run = 0;
    for (int v = 127; v >= 0; --v) { basesh[v] = run; run += total[v]; }
  }
  __syncthreads();
  int run = basesh[s];
  for (int i = 0; i < TT; ++i) { int c = col[i]; col[i] = run; run += c; }
}

// ---------------- kernel 4: init candidate slots (NEG padding) ----------------
__global__ void init_cand_kernel(int* __restrict__ cs, int* __restrict__ cl,
                                 float* __restrict__ cb) {
  int i = blockIdx.x * 256 + threadIdx.x;
  if (i >= BB*KPRE) return;
  cs[i] = NEGV; cl[i] = 0;
  cb[i*4+0] = 0.f; cb[i*4+1] = 0.f; cb[i*4+2] = 0.f; cb[i*4+3] = 0.f;
}

// ---------------- kernel 5: stable gather + deferred box decode ----------------
__global__ void gather_decode_kernel(const int* __restrict__ data,
                                     const float* __restrict__ anchor,
                                     const int* __restrict__ exp_table,
                                     const int* __restrict__ exp_shift_p,
                                     const int* __restrict__ score,
                                     const int* __restrict__ label,
                                     int* __restrict__ thist,
                                     int* __restrict__ cand_score,
                                     int* __restrict__ cand_label,
                                     float* __restrict__ cand_box) {
  int b = blockIdx.x, t = threadIdx.x;                // 1024 threads
  int es = exp_shift_p[0];
  float esc = 1.0f / (float)(1 << es);
  const float inv = 1.0f / 16.0f;                     // 1/2^INPUT_SHIFT
  for (int e = 0; e < CHUNK; ++e) {
    int n = t*CHUNK + e;                              // anchor index, ascending
    int s = score[(size_t)b*NNP + n];
    if (s < 1) continue;                              // box_filter_threshold
    size_t hidx = ((size_t)(b*128 + s))*TT + t;
    int pos = thist[hidx];
    thist[hidx] = pos + 1;                            // stable rank within (score)
    if (pos >= KPRE) continue;
    int a = n / HWP, pix = n - a*HWP;
    const int* dp = data + ((size_t)(b*AAN + a)*CHN)*HWP + pix;
    int d0 = dp[0], d1 = dp[HWP], d2 = dp[2*HWP], d3 = dp[3*HWP];
    const float* ap = anchor + ((size_t)a*4)*HWP + pix;
    float x1 = ap[0], y1 = ap[HWP], x2 = ap[2*HWP], y2 = ap[3*HWP];
    float aw = x2 - x1, ah = y2 - y1;
    float acx = (x1 + x2) * 0.5f, acy = (y1 + y2) * 0.5f;
    float dx = (float)d0 * inv, dy = (float)d1 * inv;
    int i2 = d2 < -128 ? -128 : (d2 > 127 ? 127 : d2);
    int i3 = d3 < -128 ? -128 : (d3 > 127 ? 127 : d3);
    float ew = (float)exp_table[i2 + 128] * esc;
    float eh = (float)exp_table[i3 + 128] * esc;
    float pcx = acx + dx * aw, pcy = acy + dy * ah;
    float pw = aw * ew, ph = ah * eh;
    float bx1 = pcx - pw * 0.5f, by1 = pcy - ph * 0.5f;
    float bx2 = pcx + pw * 0.5f, by2 = pcy + ph * 0.5f;
    bx1 = fminf(fmaxf(bx1, 0.f), IMGF); by1 = fminf(fmaxf(by1, 0.f), IMGF);
    bx2 = fminf(fmaxf(bx2, 0.f), IMGF); by2 = fminf(fmaxf(by2, 0.f), IMGF);
    size_t o = (size_t)b*KPRE + pos;
    cand_box[o*4+0] = bx1; cand_box[o*4+1] = by1;
    cand_box[o*4+2] = bx2; cand_box[o*4+3] = by2;
    cand_score[o] = s;
    cand_label[o] = label[(size_t)b*NNP + n];
  }
}

// ---------------- kernel 6: IoU suppression bitmask (async LDS column tiles) ----------------
__global__ void mask_kernel(const float* __restrict__ cand_box,
                            const int* __restrict__ cand_label,
                            unsigned* __restrict__ masks) {
  __shared__ __align__(16) float cb[32*4];
  __shared__ int clab[32];
  int b = blockIdx.z, cw = blockIdx.y;
  int row = blockIdx.x * 256 + threadIdx.x;
  int t = threadIdx.x;
  if (t < 32) {
    int j = cw*32 + t;
    if (j < KPRE) {
      async_copy_b128(&cand_box[((size_t)b*KPRE + j)*4], &cb[t*4]);  // async-to-LDS path
      clab[t] = cand_label[(size_t)b*KPRE + j];
    } else {
      cb[t*4+0] = 0.f; cb[t*4+1] = 0.f; cb[t*4+2] = 0.f; cb[t*4+3] = 0.f;
      clab[t] = -1;
    }
  }
  wait_async0();
  __syncthreads();
  if (row >= KPRE) return;
  size_t ro = ((size_t)b*KPRE + row)*4;
  float rx1 = cand_box[ro+0], ry1 = cand_box[ro+1];
  float rx2 = cand_box[ro+2], ry2 = cand_box[ro+3];
  int rlab = cand_label[(size_t)b*KPRE + row];
  float rarea = fmaxf(rx2 - rx1, 0.f) * fmaxf(ry2 - ry1, 0.f);
  unsigned m = 0;
  #pragma unroll 8
  for (int c = 0; c < 32; ++c) {
    if (clab[c] != rlab) continue;
    float cx1 = cb[c*4+0], cy1 = cb[c*4+1], cx2 = cb[c*4+2], cy2 = cb[c*4+3];
    float carea = fmaxf(cx2 - cx1, 0.f) * fmaxf(cy2 - cy1, 0.f);
    float iw = fmaxf(fminf(rx2, cx2) - fmaxf(rx1, cx1), 0.f);
    float ih = fmaxf(fminf(ry2, cy2) - fmaxf(ry1, cy1), 0.f);
    float inter = iw * ih;
    float iou = inter / fmaxf(rarea + carea - inter, 1e-6f);
    if (iou > NMSTHR) m |= (1u << c);
  }
  masks[((size_t)b*KPRE + row)*KWRD + cw] = m;
}

// ---------------- kernel 7: serial greedy reduce, one wave32 per batch ----------------
__global__ void nms_reduce_kernel(const unsigned* __restrict__ masks,
                                  const int* __restrict__ cand_score,
                                  int* __restrict__ keep) {
  int b = blockIdx.x, lane = threadIdx.x;             // 32 threads
  unsigned r0 = 0, r1 = 0, r2 = 0, r3 = 0;            // 128 removal words in VGPRs
  for (int i = 0; i < KPRE; ++i) {
    int w = i >> 5;
    int reg = w >> 5, ol = w & 31;
    unsigned rs = (reg == 0) ? r0 : (reg == 1) ? r1 : (reg == 2) ? r2 : r3;
    unsigned word = (unsigned)__shfl((int)rs, ol, 32);
    int sc = cand_score[(size_t)b*KPRE + i];
    int ki = (((word >> (i & 31)) & 1u) == 0u) && (sc > NEGV);
    if (ki) {
      const unsigned* mrow = masks + ((size_t)b*KPRE + i)*KWRD;
      r0 |= mrow[lane]; r1 |= mrow[lane + 32]; r2 |= mrow[lane + 64]; r3 |= mrow[lane + 96];
    }
    if (lane == 0) keep[(size_t)b*KPRE + i] = ki;
  }
}

// ---------------- kernel 8: stable compaction + NEG fill + output ----------------
__global__ void finalize_kernel(const int* __restrict__ keep,
                                const int* __restrict__ cand_score,
                                const float* __restrict__ cand_box,
                                const int* __restrict__ cand_label,
                                float* __restrict__ out) {
  __shared__ int sums[TT];
  int b = blockIdx.x, t = threadIdx.x;                // 1024 threads, 4 elems each
  int k[4]; int s = 0;
  for (int e = 0; e < 4; ++e) {
    int idx = t*4 + e;
    k[e] = (idx < KPRE) ? keep[(size_t)b*KPRE + idx] : 0;
    s += k[e];
  }
  sums[t] = s;
  __syncthreads();
  for (int off = 1; off < TT; off <<= 1) {            // inclusive Hillis-Steele scan
    int v = (t >= off) ? sums[t - off] : 0;
    __syncthreads();
    sums[t] += v;
    __syncthreads();
  }
  int run = sums[t] - s;                              // kept count before this chunk
  int tk = sums[TT - 1];                              // total kept this batch
  float* ob = out;                                    // [B][300][4]
  float* os = out + BB*POSTK*4;                       // [B][300]
  float* ol = out + BB*POSTK*4 + BB*POSTK;            // [B][300]
  for (int e = 0; e < 4; ++e) {
    int idx = t*4 + e;
    if (idx >= KPRE) break;
    int kp = k[e];
    int slot = kp ? run : (tk + (idx - run));         // stable: kept first, then NEG ties
    run += kp;
    if (slot < POSTK) {
      size_t ci = (size_t)b*KPRE + idx;
      size_t oo = ((size_t)b*POSTK + slot)*4;
      ob[oo+0] = rintf(cand_box[ci*4+0] * 4.0f);      // round-half-even like jnp.round
      ob[oo+1] = rintf(cand_box[ci*4+1] * 4.0f);
      ob[oo+2] = rintf(cand_box[ci*4+2] * 4.0f);
      ob[oo+3] = rintf(cand_box[ci*4+3] * 4.0f);
      os[(size_t)b*POSTK + slot] = kp ? (float)cand_score[ci] : (float)NEGV;
      ol[(size_t)b*POSTK + slot] = (float)cand_label[ci];
    }
  }
}

// ---------------- host launcher ----------------
extern "C" void kernel_launch(void* const* d_in, const int* in_sizes, int n_in,
                              void* d_out, int out_size, void* d_ws, size_t ws_size,
                              hipStream_t stream) {
  (void)in_sizes; (void)n_in; (void)out_size; (void)ws_size;
  const int*   data      = (const int*)d_in[0];
  const float* anchor    = (const float*)d_in[1];
  const int*   exp_table = (const int*)d_in[2];
  const int*   exp_shift = (const int*)d_in[3];
  float* out = (float*)d_out;

  char* ws = (char*)d_ws;
  size_t o = 0;
  auto carve = [&](size_t bytes) -> char* {
    char* p = ws + o;
    o = (o + bytes + 255) & ~(size_t)255;
    return p;
  };
  int*      score      = (int*)     carve((size_t)BB*NNP*4);
  int*      label      = (int*)     carve((size_t)BB*NNP*4);
  int*      thist      = (int*)     carve((size_t)BB*128*TT*4);
  int*      cand_score = (int*)     carve((size_t)BB*KPRE*4);
  int*      cand_label = (int*)     carve((size_t)BB*KPRE*4);
  float*    cand_box   = (float*)   carve((size_t)BB*KPRE*16);
  unsigned* masks      = (unsigned*)carve((size_t)BB*KPRE*KWRD*4);
  int*      keep       = (int*)     carve((size_t)BB*KPRE*4);

  score_kernel<<<dim3(HWP/4/256, AAN, BB), 256, 0, stream>>>(data, score, label);
  hist_kernel<<<BB, TT, 0, stream>>>(score, thist);
  scan_kernel<<<BB, 128, 0, stream>>>(thist);
  init_cand_kernel<<<(BB*KPRE + 255)/256, 256, 0, stream>>>(cand_score, cand_label, cand_box);
  gather_decode_kernel<<<BB, TT, 0, stream>>>(data, anchor, exp_table, exp_shift,
                                              score, label, thist,
                                              cand_score, cand_label, cand_box);
  mask_kernel<<<dim3((KPRE + 255)/256, KWRD, BB), 256, 0, stream>>>(cand_box, cand_label, masks);
  nms_reduce_kernel<<<BB, 32, 0, stream>>>(masks, cand_score, keep);
  finalize_kernel<<<BB, TT, 0, stream>>>(keep, cand_score, cand_box, cand_label, out);
}